// DeltaNet_70317204570457
// MI455X (gfx1250) — compile-verified
//
#include <hip/hip_runtime.h>
#include <hip/hip_bf16.h>
#include <math.h>

typedef __attribute__((ext_vector_type(2))) float v2f;
typedef __attribute__((ext_vector_type(8))) float v8f;
typedef __attribute__((ext_vector_type(4))) unsigned int v4u;
typedef __attribute__((ext_vector_type(8))) int v8i;
typedef __attribute__((ext_vector_type(4))) int v4i;

__device__ __forceinline__ v8f wmma_f32x4(v2f a, v2f b, v8f c) {
    // D = A(16x4 f32) * B(4x16 f32) + C(16x16 f32)
    return __builtin_amdgcn_wmma_f32_16x16x4_f32(
        /*neg_a=*/false, a, /*neg_b=*/false, b,
        /*c_mod=*/(short)0, c, /*reuse_a=*/false, /*reuse_b=*/false);
}

// ---------------------------------------------------------------------------
// Tensor Data Mover: 2D tile load (row-major, 4-byte elements) into LDS.
// Builds D# group0/group1 per CDNA5 ISA 8.3/8.4; groups 2/3 zero (2D tensor).
// Issue once per workgroup (from one wave); tracked with TENSORcnt.
// This toolchain exposes the 6-arg builtin: (g0, g1, g2, g3, g_extra, cpol).
// ---------------------------------------------------------------------------
__device__ __forceinline__ void tdm_load_2d(const void* gptr, unsigned lds_off_bytes,
                                            unsigned tile_w_elems, unsigned tile_h_rows,
                                            unsigned row_stride_elems) {
    unsigned long long ga = (unsigned long long)(uintptr_t)gptr;
    v4u g0;
    g0.x = 1u;                                             // count=1 (valid user D#)
    g0.y = lds_off_bytes;                                  // lds_addr
    g0.z = (unsigned)(ga & 0xFFFFFFFFull);                 // global_addr[31:0]
    g0.w = (unsigned)((ga >> 32) & 0x01FFFFFFull)          // global_addr[56:32]
         | 0x80000000u;                                    // type=2 ("image")
    const unsigned td0 = 0x7FFFFFFFu, td1 = 0x7FFFFFFFu;   // huge dims: no OOB clip
    v8i g1;
    g1[0] = (int)(2u << 16);                               // data_size=2 (4 bytes)
    g1[1] = (int)((td0 & 0xFFFFu) << 16);                  // tensor_dim0[15:0]
    g1[2] = (int)((td0 >> 16) | ((td1 & 0xFFFFu) << 16));  // td0[31:16] | td1[15:0]
    g1[3] = (int)((td1 >> 16) | (tile_w_elems << 16));     // td1[31:16] | tile_dim0
    g1[4] = (int)tile_h_rows;                              // tile_dim1 (tile_dim2=0)
    g1[5] = (int)row_stride_elems;                         // tensor_dim0_stride[31:0]
    g1[6] = 0;                                             // stride hi / dim1_stride lo
    g1[7] = 0;
    v4i zz4 = {};
    v8i zz8 = {};
    __builtin_amdgcn_tensor_load_to_lds(g0, g1, zz4, zz4, zz8, 0);
}

// ---------------------------------------------------------------------------
// Wide GEMM: Y[M,N] = A[M,Kd] @ W[N,Kd]^T (+bias). Requires M%128==0, N%128==0.
// Block = 256 threads = 8 waves arranged 4(M) x 2(N); each wave computes a
// 32x64 strip = 2x4 WMMA tiles (8 accumulators). No divergent guards anywhere.
// ---------------------------------------------------------------------------
__global__ void gemm_nt_wmma_wide(const float* __restrict__ A, const float* __restrict__ W,
                                  const float* __restrict__ bias, float* __restrict__ Y,
                                  int M, int N, int Kd) {
    int wave = threadIdx.x >> 5;
    int lane = threadIdx.x & 31;
    int wm = wave & 3;                   // 0..3
    int wn = wave >> 2;                  // 0..1
    int m0 = blockIdx.x * 128 + wm * 32;
    int n0 = blockIdx.y * 128 + wn * 64;
    int lr = lane & 15;
    int hi = lane >> 4;

    const float* a0p = A + (size_t)(m0 + lr) * Kd;
    const float* a1p = A + (size_t)(m0 + 16 + lr) * Kd;
    const float* wp[4];
#pragma unroll
    for (int t = 0; t < 4; ++t) wp[t] = W + (size_t)(n0 + t * 16 + lr) * Kd;

    v8f acc[2][4];
#pragma unroll
    for (int mi = 0; mi < 2; ++mi)
#pragma unroll
        for (int t = 0; t < 4; ++t) acc[mi][t] = (v8f){};

    for (int kk = 0; kk < Kd; kk += 4) {
        int ka = kk + hi * 2;
        v2f a0, a1;
        a0.x = a0p[ka]; a0.y = a0p[ka + 1];
        a1.x = a1p[ka]; a1.y = a1p[ka + 1];
        v2f b[4];
#pragma unroll
        for (int t = 0; t < 4; ++t) { b[t].x = wp[t][ka]; b[t].y = wp[t][ka + 1]; }
#pragma unroll
        for (int t = 0; t < 4; ++t) acc[0][t] = wmma_f32x4(a0, b[t], acc[0][t]);
#pragma unroll
        for (int t = 0; t < 4; ++t) acc[1][t] = wmma_f32x4(a1, b[t], acc[1][t]);
    }

#pragma unroll
    for (int mi = 0; mi < 2; ++mi)
#pragma unroll
        for (int t = 0; t < 4; ++t)
#pragma unroll
            for (int r = 0; r < 8; ++r) {
                int m = m0 + mi * 16 + r + hi * 8;
                int n = n0 + t * 16 + lr;
                float val = acc[mi][t][r];
                if (bias) val += bias[n];
                Y[(size_t)m * N + n] = val;
            }
}

// ---------------------------------------------------------------------------
// Small-N GEMM (gate N=16, alpha N=8): one 16x16 tile per wave, branch-free
// B-operand guard (clamped row + mask multiply -> selects, no EXEC writes).
// ---------------------------------------------------------------------------
__global__ void gemm_nt_wmma_small(const float* __restrict__ A, const float* __restrict__ W,
                                   const float* __restrict__ bias, float* __restrict__ Y,
                                   int M, int N, int Kd, int apply_sigmoid) {
    int wave = threadIdx.x >> 5;
    int lane = threadIdx.x & 31;
    int m0 = (blockIdx.x * 8 + wave) * 16;
    int lr = lane & 15;
    int hi = lane >> 4;
    int wrow = lr;                       // n0 == 0 (single tile of columns)
    int wrc = (wrow < N) ? wrow : 0;
    float msk = (wrow < N) ? 1.f : 0.f;

    const float* ap = A + (size_t)(m0 + lr) * Kd;
    const float* wpr = W + (size_t)wrc * Kd;

    v8f acc = {};
    for (int kk = 0; kk < Kd; kk += 4) {
        int ka = kk + hi * 2;
        v2f a, b;
        a.x = ap[ka];           a.y = ap[ka + 1];
        b.x = wpr[ka] * msk;    b.y = wpr[ka + 1] * msk;
        acc = wmma_f32x4(a, b, acc);
    }
#pragma unroll
    for (int r = 0; r < 8; ++r) {
        int m = m0 + r + hi * 8;
        int n = lr;
        if (n < N) {
            float val = acc[r];
            if (bias) val += bias[n];
            if (apply_sigmoid) val = 1.f / (1.f + __expf(-val));
            Y[(size_t)m * N + n] = val;
        }
    }
}

// ---------------------------------------------------------------------------
// Multi-timescale gated linear attention recurrence.
// grid = B*H blocks, block = K*dh = 128 threads.
// Thread (kk, j) owns state column Hm[b,h,kk,:,j] in registers (64 floats).
// ---------------------------------------------------------------------------
__global__ void linatt_kernel(const float* __restrict__ q, const float* __restrict__ k,
                              const float* __restrict__ v, const float* __restrict__ gate,
                              const float* __restrict__ decay_logit,
                              float* __restrict__ y_lin, int B, int T, int H, int Kts) {
    const int dh = 64;
    int b = blockIdx.x / H;
    int h = blockIdx.x % H;
    int kk = threadIdx.x / dh;   // timescale index
    int j  = threadIdx.x % dh;   // state column

    float lam = 1.f / (1.f + __expf(-decay_logit[h * Kts + kk]));

    float Hcol[64];
#pragma unroll
    for (int i = 0; i < 64; ++i) Hcol[i] = 0.f;
    float Z = 0.f;

    __shared__ float sq[64];
    __shared__ float sk[64];
    __shared__ float sy[128];

    for (int t = 0; t < T; ++t) {
        size_t base = (((size_t)b * T + t) * H + h) * dh;
        if (threadIdx.x < 64) sq[threadIdx.x] = q[base + threadIdx.x];
        else                  sk[threadIdx.x - 64] = k[base + threadIdx.x - 64];
        __syncthreads();

        float g = gate[(((size_t)b * T + t) * H + h) * Kts + kk];
        float s = 0.f;
#pragma unroll
        for (int i = 0; i < 64; ++i) s += fabsf(sk[i]);
        s = s * (1.f / 64.f) + 1e-6f;
        Z = Z * lam + g * s;

        float vj = v[base + j];
        float acc = 0.f;
#pragma unroll
        for (int i = 0; i < 64; ++i) {
            Hcol[i] = Hcol[i] * lam + g * sk[i] * vj;
            acc += sq[i] * Hcol[i];
        }
        sy[threadIdx.x] = acc / (Z + 1e-6f);
        __syncthreads();
        if (threadIdx.x < 64)
            y_lin[base + threadIdx.x] = sy[threadIdx.x] + sy[64 + threadIdx.x];
        __syncthreads();
    }
}

// ---------------------------------------------------------------------------
// Sliding-window local attention with ALiBi, one block per (b, chunk, head).
// Block = 256 threads (8 waves). Scores and P@V via fp32 WMMA.
// Q/K/V tiles staged into LDS by the Tensor Data Mover (TENSORcnt-tracked);
// manual zero-filled staging only for the c==0 boundary chunk.
// LDS: phase1 Q[64x64]+K[128x64] (aliased as S[64][129] in phase2) + V[128x64].
// ---------------------------------------------------------------------------
__global__ void local_attn_wmma(const float* __restrict__ q, const float* __restrict__ k,
                                const float* __restrict__ v, float* __restrict__ y_loc,
                                int B, int T, int H) {
    const int dh = 64;
    const int W = 64;
    const int SLD = 129;
    int nc = T / W;
    int h = blockIdx.x % H;
    int c = (blockIdx.x / H) % nc;
    int b = blockIdx.x / (H * nc);

    __shared__ float smem[64 * 64 + 128 * 64];   // Q | K, aliased as S after use
    __shared__ float sV[128 * 64];
    float* sQ = smem;
    float* sK = smem + 64 * 64;
    float* sS = smem;                            // aliased after barrier

    int tid = threadIdx.x;

    // ---- stage tiles via TDM (one issuing wave) ----
    const float* qg = q + (((size_t)b * T + c * W) * H + h) * dh;
    if (tid < 32) {
        tdm_load_2d(qg, (unsigned)(uintptr_t)(void*)sQ, 64, 64, H * dh);
        if (c > 0) {
            const float* kg = k + (((size_t)b * T + (c - 1) * W) * H + h) * dh;
            const float* vg = v + (((size_t)b * T + (c - 1) * W) * H + h) * dh;
            tdm_load_2d(kg, (unsigned)(uintptr_t)(void*)sK, 64, 128, H * dh);
            tdm_load_2d(vg, (unsigned)(uintptr_t)(void*)sV, 64, 128, H * dh);
        }
        __builtin_amdgcn_s_wait_tensorcnt(0);
    }
    if (c == 0) {
        // boundary chunk: prev-chunk rows are zero, cur-chunk rows are tokens 0..63
        for (int idx = tid; idx < 128 * 64; idx += 256) {
            int jj = idx >> 6, d = idx & 63;
            float kvl = 0.f, vvl = 0.f;
            if (jj >= 64) {
                size_t g = (((size_t)b * T + (jj - 64)) * H + h) * dh + d;
                kvl = k[g];
                vvl = v[g];
            }
            sK[idx] = kvl;
            sV[idx] = vvl;
        }
    }
    __syncthreads();

    int wave = tid >> 5, lane = tid & 31, lr = lane & 15, hi = lane >> 4;

    // scores: 64x128 = 4x8 tiles of 16x16; wave handles 4 tiles
    v8f sc[4];
    for (int it = 0; it < 4; ++it) {
        int id = wave * 4 + it;
        int tm = id >> 3, tn = id & 7;
        v8f acc = {};
        for (int kkk = 0; kkk < 64; kkk += 4) {
            int ka = kkk + hi * 2;
            v2f a, bb;
            a.x  = sQ[(tm * 16 + lr) * 64 + ka];
            a.y  = sQ[(tm * 16 + lr) * 64 + ka + 1];
            bb.x = sK[(tn * 16 + lr) * 64 + ka];
            bb.y = sK[(tn * 16 + lr) * 64 + ka + 1];
            acc = wmma_f32x4(a, bb, acc);
        }
        sc[it] = acc;
    }
    __syncthreads();   // done reading sQ/sK; safe to alias as sS

    // ALiBi + causal window mask, write scores to LDS
    float slope = exp2f(-8.0f * (float)(h + 1) / (float)H);
    for (int it = 0; it < 4; ++it) {
        int id = wave * 4 + it;
        int tm = id >> 3, tn = id & 7;
#pragma unroll
        for (int r = 0; r < 8; ++r) {
            int i = tm * 16 + r + hi * 8;
            int jj = tn * 16 + lr;
            float val = sc[it][r];
            bool valid = (jj >= i + 1) && (jj <= i + W) && (c * W + jj - W >= 0);
            val = valid ? (val - slope * (float)(i + W - jj)) : -1e30f;
            sS[i * SLD + jj] = val;
        }
    }
    __syncthreads();

    // row softmax (threads 0..63, one row each)
    if (tid < 64) {
        float mx = -3.4e38f;
        for (int jj = 0; jj < 128; ++jj) mx = fmaxf(mx, sS[tid * SLD + jj]);
        float sum = 0.f;
        for (int jj = 0; jj < 128; ++jj) {
            float e = __expf(sS[tid * SLD + jj] - mx);
            sS[tid * SLD + jj] = e;
            sum += e;
        }
        float inv = 1.f / sum;
        for (int jj = 0; jj < 128; ++jj) sS[tid * SLD + jj] *= inv;
    }
    __syncthreads();

    // y = P[64,128] @ Vctx[128,64]; 16 tiles, wave handles 2; V from LDS
    for (int it = 0; it < 2; ++it) {
        int id = wave * 2 + it;
        int tm = id >> 2, tn = id & 3;
        int n = tn * 16 + lr;
        v8f acc = {};
        for (int kkk = 0; kkk < 128; kkk += 4) {
            int ka = kkk + hi * 2;
            v2f a, bb;
            a.x = sS[(tm * 16 + lr) * SLD + ka];
            a.y = sS[(tm * 16 + lr) * SLD + ka + 1];
            bb.x = sV[ka * 64 + n];
            bb.y = sV[(ka + 1) * 64 + n];
            acc = wmma_f32x4(a, bb, acc);
        }
#pragma unroll
        for (int r = 0; r < 8; ++r) {
            int i = tm * 16 + r + hi * 8;
            y_loc[(((size_t)b * T + c * W + i) * H + h) * dh + n] = acc[r];
        }
    }
}

// ---------------------------------------------------------------------------
// y = alpha * y_loc + (1 - alpha) * y_lin   (alpha per [B,T,H])
// ---------------------------------------------------------------------------
__global__ void mix_kernel(const float* __restrict__ y_loc, const float* __restrict__ y_lin,
                           const float* __restrict__ alpha, float* __restrict__ y,
                           size_t total, int dh) {
    size_t idx = (size_t)blockIdx.x * blockDim.x + threadIdx.x;
    if (idx < total) {
        size_t bth = idx / dh;     // (b*T + t)*H + h
        float a = alpha[bth];
        y[idx] = a * y_loc[idx] + (1.f - a) * y_lin[idx];
    }
}

// ---------------------------------------------------------------------------
// out = LayerNorm(x + proj) * gamma + beta ; block per (b,t) row.
// ---------------------------------------------------------------------------
__global__ void addnorm_kernel(const float* __restrict__ x, const float* __restrict__ proj,
                               const float* __restrict__ gamma, const float* __restrict__ beta,
                               float* __restrict__ out, int D) {
    __shared__ float red[256];
    size_t row = blockIdx.x;
    const float* xr = x + row * D;
    const float* pr = proj + row * D;

    float s = 0.f, s2 = 0.f;
    for (int d = threadIdx.x; d < D; d += 256) {
        float r = xr[d] + pr[d];
        s += r; s2 += r * r;
    }
    red[threadIdx.x] = s;
    __syncthreads();
    for (int off = 128; off > 0; off >>= 1) {
        if (threadIdx.x < off) red[threadIdx.x] += red[threadIdx.x + off];
        __syncthreads();
    }
    float mean = red[0] / (float)D;
    __syncthreads();
    red[threadIdx.x] = s2;
    __syncthreads();
    for (int off = 128; off > 0; off >>= 1) {
        if (threadIdx.x < off) red[threadIdx.x] += red[threadIdx.x + off];
        __syncthreads();
    }
    float var = red[0] / (float)D - mean * mean;
    float inv = rsqrtf(var + 1e-5f);
    for (int d = threadIdx.x; d < D; d += 256) {
        float r = xr[d] + pr[d];
        out[row * D + d] = (r - mean) * inv * gamma[d] + beta[d];
    }
}

// ---------------------------------------------------------------------------
extern "C" void kernel_launch(void* const* d_in, const int* in_sizes, int n_in,
                              void* d_out, int out_size, void* d_ws, size_t ws_size,
                              hipStream_t stream) {
    const int D = 512, H = 8, Kts = 2, dh = 64, T = 2048;
    const int M = in_sizes[0] / D;          // B*T = 4096
    const int B = M / T;                    // 2
    const int nc = T / 64;

    const float* x      = (const float*)d_in[0];
    const float* Wq     = (const float*)d_in[1];
    const float* Wk     = (const float*)d_in[2];
    const float* Wv     = (const float*)d_in[3];
    const float* Wg     = (const float*)d_in[4];
    const float* bg     = (const float*)d_in[5];
    const float* Wa     = (const float*)d_in[6];
    const float* ba     = (const float*)d_in[7];
    const float* Wo     = (const float*)d_in[8];
    const float* bo     = (const float*)d_in[9];
    const float* decay  = (const float*)d_in[10];
    const float* gamma  = (const float*)d_in[11];
    const float* beta   = (const float*)d_in[12];
    float* out = (float*)d_out;

    // workspace carve-up
    float* ws = (float*)d_ws;
    size_t off = 0;
    float* q_buf    = ws + off; off += (size_t)M * D;
    float* k_buf    = ws + off; off += (size_t)M * D;
    float* v_buf    = ws + off; off += (size_t)M * D;
    float* gate_buf = ws + off; off += (size_t)M * H * Kts;
    float* alp_buf  = ws + off; off += (size_t)M * H;
    float* ylin_buf = ws + off; off += (size_t)M * D;
    float* yloc_buf = ws + off; off += (size_t)M * D;
    float* ymix_buf = ws + off; off += (size_t)M * D;
    float* proj_buf = ws + off; off += (size_t)M * D;

    dim3 blk(256);
    dim3 grid_wide(M / 128, D / 128);       // 32 x 4
    dim3 grid_small(M / 128, 1);

    // projections (wide, guard-free WMMA GEMMs)
    gemm_nt_wmma_wide<<<grid_wide, blk, 0, stream>>>(x, Wq, nullptr, q_buf, M, D, D);
    gemm_nt_wmma_wide<<<grid_wide, blk, 0, stream>>>(x, Wk, nullptr, k_buf, M, D, D);
    gemm_nt_wmma_wide<<<grid_wide, blk, 0, stream>>>(x, Wv, nullptr, v_buf, M, D, D);
    // skinny projections with sigmoid fused
    gemm_nt_wmma_small<<<grid_small, blk, 0, stream>>>(x, Wg, bg, gate_buf, M, H * Kts, D, 1);
    gemm_nt_wmma_small<<<grid_small, blk, 0, stream>>>(x, Wa, ba, alp_buf,  M, H,       D, 1);

    // serial gated linear attention
    linatt_kernel<<<B * H, 128, 0, stream>>>(q_buf, k_buf, v_buf, gate_buf, decay,
                                             ylin_buf, B, T, H, Kts);

    // sliding-window local attention with ALiBi (TDM-staged tiles)
    local_attn_wmma<<<B * nc * H, 256, 0, stream>>>(q_buf, k_buf, v_buf, yloc_buf, B, T, H);

    // gated blend
    size_t total = (size_t)M * D;
    mix_kernel<<<(unsigned)((total + 255) / 256), blk, 0, stream>>>(yloc_buf, ylin_buf,
                                                                    alp_buf, ymix_buf,
                                                                    total, dh);

    // output projection + residual + layernorm
    gemm_nt_wmma_wide<<<grid_wide, blk, 0, stream>>>(ymix_buf, Wo, bo, proj_buf, M, D, D);
    addnorm_kernel<<<M, blk, 0, stream>>>(x, proj_buf, gamma, beta, out, D);
}